// DiscriminativeLoss_35948876268437
// MI455X (gfx1250) — compile-verified
//
#include <hip/hip_runtime.h>

typedef __attribute__((ext_vector_type(2))) float v2f;
typedef __attribute__((ext_vector_type(8))) float v8f;

// Workspace layout (floats):
//   [0 .. 511]   gsums   (32 clusters x 16 feats)
//   [512.. 543]  gcnt    (32)
//   [544.. 575]  ghinge  (32)
//   [576]        gpartial (beta*dist_loss + gamma*reg)
//   [577..1088]  gmeans  (32 x 16)
#define WS_SUMS   0
#define WS_CNT    512
#define WS_HINGE  544
#define WS_PART   576
#define WS_MEANS  577

__global__ void k_init(float* __restrict__ ws) {
    int i = blockIdx.x * blockDim.x + threadIdx.x;
    if (i < 577) ws[i] = 0.0f;   // gmeans fully overwritten later; no need to zero
}

// ---------------------------------------------------------------------------
// Pass 1: segment sum via one-hot WMMA.
//   sums[32x16] += H[32 x 4pts] * X[4pts x 16] per WMMA group,
//   using V_WMMA_F32_16X16X4_F32 (cluster tiles 0-15 and 16-31).
// Unrolled x2: 8 points / wave-iteration with two independent accumulator
// pairs so the WMMA accumulate chains and the global loads overlap.
// A (one-hot) layout: lane<16 -> (M=lane, K=0/1), lane>=16 -> (M=lane-16, K=2/3)
// B (x data) layout:  lane<16 -> (K=0/1, N=lane), lane>=16 -> (K=2/3, N=lane-16)
// => lane loads x[base + 2*half + {0,1}][lane&15]  (fully coalesced 512B/iter)
// ---------------------------------------------------------------------------
__global__ __launch_bounds__(256) void k_pass1(const float* __restrict__ x,
                                               const int* __restrict__ ylo,
                                               int N, float* __restrict__ ws) {
    float* gsums = ws + WS_SUMS;
    float* gcnt  = ws + WS_CNT;
    __shared__ float lsums[512];
    __shared__ float lcnt[32];
    for (int i = threadIdx.x; i < 512; i += blockDim.x) lsums[i] = 0.0f;
    if (threadIdx.x < 32) lcnt[threadIdx.x] = 0.0f;
    __syncthreads();

    const int lane = threadIdx.x & 31;
    const int half = lane >> 4;       // 0: points {0,1}, 1: points {2,3} of a group
    const int l15  = lane & 15;
    const long long wid = (long long)((blockIdx.x * blockDim.x + threadIdx.x) >> 5);
    const long long nw  = (long long)((gridDim.x * blockDim.x) >> 5);

    v8f c0a = {}, c1a = {};            // group A: clusters 0..15 / 16..31
    v8f c0b = {}, c1b = {};            // group B (unroll partner)
    float cnt0 = 0.0f, cnt1 = 0.0f;
    const long long Nfull = (long long)(N & ~7);
    const int xoff = half * 32 + l15;  // within a 4-point group
    const int pA   = half << 1;

    for (long long base = wid * 8; base < Nfull; base += nw * 8) {
        const float* xb = x + base * 16;
        const int* yb = ylo + 2 * base;
        // ---- group A: points base .. base+3 ----
        v2f bA; bA.x = xb[xoff]; bA.y = xb[xoff + 16];
        const int la0 = yb[2 * pA];
        const int la1 = yb[2 * pA + 2];
        // ---- group B: points base+4 .. base+7 ----
        v2f bB; bB.x = xb[64 + xoff]; bB.y = xb[64 + xoff + 16];
        const int lb0 = yb[8 + 2 * pA];
        const int lb1 = yb[8 + 2 * pA + 2];

        v2f a0, a1;
        a0.x = (la0 == l15)      ? 1.0f : 0.0f;
        a0.y = (la1 == l15)      ? 1.0f : 0.0f;
        a1.x = (la0 == l15 + 16) ? 1.0f : 0.0f;
        a1.y = (la1 == l15 + 16) ? 1.0f : 0.0f;
        cnt0 += a0.x + a0.y;
        cnt1 += a1.x + a1.y;
        c0a = __builtin_amdgcn_wmma_f32_16x16x4_f32(false, a0, false, bA,
                                                    (short)0, c0a, false, false);
        c1a = __builtin_amdgcn_wmma_f32_16x16x4_f32(false, a1, false, bA,
                                                    (short)0, c1a, false, false);

        v2f d0, d1;
        d0.x = (lb0 == l15)      ? 1.0f : 0.0f;
        d0.y = (lb1 == l15)      ? 1.0f : 0.0f;
        d1.x = (lb0 == l15 + 16) ? 1.0f : 0.0f;
        d1.y = (lb1 == l15 + 16) ? 1.0f : 0.0f;
        cnt0 += d0.x + d0.y;
        cnt1 += d1.x + d1.y;
        c0b = __builtin_amdgcn_wmma_f32_16x16x4_f32(false, d0, false, bB,
                                                    (short)0, c0b, false, false);
        c1b = __builtin_amdgcn_wmma_f32_16x16x4_f32(false, d1, false, bB,
                                                    (short)0, c1b, false, false);
    }

    const v8f c0 = c0a + c0b;
    const v8f c1 = c1a + c1b;

    // Flush per-wave accumulators: C tile VGPR e, lane<16 -> cluster e, lane>=16 -> cluster e+8
#pragma unroll
    for (int e = 0; e < 8; ++e) {
        const int cl = e + (half << 3);
        atomicAdd(&lsums[cl * 16 + l15], c0[e]);
        atomicAdd(&lsums[(16 + cl) * 16 + l15], c1[e]);
    }
    atomicAdd(&lcnt[l15], cnt0);
    atomicAdd(&lcnt[16 + l15], cnt1);
    __syncthreads();

    for (int i = threadIdx.x; i < 512; i += blockDim.x) atomicAdd(&gsums[i], lsums[i]);
    if (threadIdx.x < 32) atomicAdd(&gcnt[threadIdx.x], lcnt[threadIdx.x]);

    // Tail (N % 8 points), negligible work
    if (blockIdx.x == 0 && threadIdx.x < (N & 7)) {
        const long long p = Nfull + threadIdx.x;
        const int l = ylo[2 * p];
        atomicAdd(&gcnt[l], 1.0f);
        for (int f = 0; f < 16; ++f)
            atomicAdd(&gsums[l * 16 + f], x[p * 16 + f]);
    }
}

// ---------------------------------------------------------------------------
// Means + pairwise loss: Gram matrix G = M * M^T (32x32x16) via 16 fp32 WMMAs,
// then dist_loss from d2(i,j) = G[i][i] + G[j][j] - 2 G[i][j], reg from diag.
// ---------------------------------------------------------------------------
__global__ __launch_bounds__(256) void k_pairwise(float* __restrict__ ws) {
    const float* gsums = ws + WS_SUMS;
    const float* gcnt  = ws + WS_CNT;
    float* gpart  = ws + WS_PART;
    float* gmeans = ws + WS_MEANS;
    __shared__ __align__(16) float lm[512];
    __shared__ float lg[1024];        // G, 32x32
    __shared__ float red[2];          // {pair hinge^2 sum, reg sum}

    const int tid = threadIdx.x;
    if (tid == 0) { red[0] = 0.0f; red[1] = 0.0f; }
    for (int i = tid; i < 512; i += blockDim.x) {
        const float c = gcnt[i >> 4];
        const float m = gsums[i] / fmaxf(c, 1.0f);
        lm[i] = m;
        gmeans[i] = m;
    }
    __syncthreads();

    if (tid < 32) {                    // wave 0, EXEC all ones
        const int half = tid >> 4;
        const int l15  = tid & 15;
        // mat[t][s]: serves as both A (rows 16t..16t+15) and B (cols) for K-step s
        v2f mat[2][4];
#pragma unroll
        for (int t = 0; t < 2; ++t)
#pragma unroll
            for (int s = 0; s < 4; ++s) {
                const int idx = (16 * t + l15) * 16 + 4 * s + 2 * half;
                mat[t][s].x = lm[idx];
                mat[t][s].y = lm[idx + 1];
            }
#pragma unroll
        for (int I = 0; I < 2; ++I)
#pragma unroll
            for (int J = 0; J < 2; ++J) {
                v8f acc = {};
#pragma unroll
                for (int s = 0; s < 4; ++s)
                    acc = __builtin_amdgcn_wmma_f32_16x16x4_f32(
                        false, mat[I][s], false, mat[J][s], (short)0, acc, false, false);
#pragma unroll
                for (int e = 0; e < 8; ++e)
                    lg[(16 * I + e + 8 * half) * 32 + 16 * J + l15] = acc[e];
            }
    }
    __syncthreads();

    float local = 0.0f;
    for (int i = tid; i < 1024; i += blockDim.x) {
        const int r = i >> 5, c = i & 31;
        if (r != c) {
            const float d2 = lg[r * 32 + r] + lg[c * 32 + c] - 2.0f * lg[r * 32 + c];
            const float pd = sqrtf(fmaxf(d2, 0.0f));
            const float h  = fmaxf(3.0f - pd, 0.0f);   // 2*DELTA_DIST = 3.0
            local += h * h;
        }
    }
    atomicAdd(&red[0], local);
    if (tid < 32) atomicAdd(&red[1], sqrtf(fmaxf(lg[tid * 32 + tid], 0.0f)));
    __syncthreads();
    if (tid == 0)
        gpart[0] = 1.0f * (red[0] / (31.0f * 32.0f)) + 0.001f * (red[1] / 32.0f);
}

// ---------------------------------------------------------------------------
// Pass 2: per-point hinge, pure streaming (bandwidth bound; means live in LDS,
// next grid-stride element prefetched -> global_prefetch_b8)
// ---------------------------------------------------------------------------
__global__ __launch_bounds__(256) void k_pass2(const float* __restrict__ x,
                                               const int* __restrict__ ylo,
                                               int N, float* __restrict__ ws) {
    float* ghinge = ws + WS_HINGE;
    const float* gmeans = ws + WS_MEANS;
    __shared__ __align__(16) float lm[512];
    __shared__ float lh[32];
    for (int i = threadIdx.x; i < 512; i += blockDim.x) lm[i] = gmeans[i];
    if (threadIdx.x < 32) lh[threadIdx.x] = 0.0f;
    __syncthreads();

    const long long stride = (long long)gridDim.x * blockDim.x;
    for (long long i = (long long)blockIdx.x * blockDim.x + threadIdx.x; i < N; i += stride) {
        if (i + stride < N) __builtin_prefetch(x + (i + stride) * 16, 0, 1);
        const float4* xp = (const float4*)(x + i * 16);
        const int l = ylo[2 * i];
        const float4* mp = (const float4*)(lm + l * 16);
        float d2 = 0.0f;
#pragma unroll
        for (int q = 0; q < 4; ++q) {
            const float4 xv = xp[q];
            const float4 mv = mp[q];
            const float dx = xv.x - mv.x, dy = xv.y - mv.y;
            const float dz = xv.z - mv.z, dw = xv.w - mv.w;
            d2 += dx * dx + dy * dy + dz * dz + dw * dw;
        }
        const float h = fmaxf(sqrtf(d2) - 1.0f, 0.0f);
        atomicAdd(&lh[l], h * h);
    }
    __syncthreads();
    if (threadIdx.x < 32) atomicAdd(&ghinge[threadIdx.x], lh[threadIdx.x]);
}

__global__ void k_final(const float* __restrict__ ws, float* __restrict__ out) {
    if (threadIdx.x == 0) {
        const float* gcnt = ws + WS_CNT;
        const float* gh   = ws + WS_HINGE;
        float v = 0.0f;
        for (int k = 0; k < 32; ++k) v += gh[k] / fmaxf(gcnt[k], 1.0f);
        out[0] = 1.0f * (v / 32.0f) + ws[WS_PART];
    }
}

extern "C" void kernel_launch(void* const* d_in, const int* in_sizes, int n_in,
                              void* d_out, int out_size, void* d_ws, size_t ws_size,
                              hipStream_t stream) {
    (void)n_in; (void)out_size; (void)ws_size;
    const float* x = (const float*)d_in[0];
    const int* ylo = (const int*)d_in[1];     // int64 labels; read low dwords
    const int N = in_sizes[0] / 16;
    float* ws = (float*)d_ws;

    k_init<<<3, 256, 0, stream>>>(ws);
    k_pass1<<<1024, 256, 0, stream>>>(x, ylo, N, ws);
    k_pairwise<<<1, 256, 0, stream>>>(ws);
    k_pass2<<<2048, 256, 0, stream>>>(x, ylo, N, ws);
    k_final<<<1, 32, 0, stream>>>(ws, (float*)d_out);
}